// SpanPruner_53523882443266
// MI455X (gfx1250) — compile-verified
//
#include <hip/hip_runtime.h>
#include <hip/hip_bf16.h>

typedef __attribute__((ext_vector_type(2))) float v2f;
typedef __attribute__((ext_vector_type(8))) float v8f;
typedef __attribute__((ext_vector_type(4))) unsigned int u32x4;
typedef __attribute__((ext_vector_type(8))) int i32x8;
typedef __attribute__((ext_vector_type(4))) int i32x4;

#define MAX_E 1024

// ---------------------------------------------------------------------------
// Kernel 1: scores = emb @ w + b + log(mask) via V_WMMA_F32_16X16X4_F32.
// One wave32 per 16 rows.
//   A layout (ISA 7.12.2, 32-bit A 16x4): lane L   = row L, K = k..k+1
//                                         lane L+16 = row L, K = k+2..k+3
//   -> loaded DIRECTLY with one per-lane global b64 (lanes L and L+16 cover
//      16 contiguous bytes of row L); no cross-lane permutes needed.
//   B (4x16) is w[k..k+3] broadcast to every column, so all 16 accumulator
//   columns carry the identical running dot product.
//   C/D layout: VGPR m = rows m (lanes 0-15) and m+8 (lanes 16-31).
// w is staged into LDS by the Tensor Data Mover (1D tile descriptor, wave 0
// only, completed with s_wait_tensorcnt) when the TDM builtin is available.
// ---------------------------------------------------------------------------
__global__ __launch_bounds__(256) void sp_score_wmma(
    const float* __restrict__ emb, const float* __restrict__ w,
    const float* __restrict__ b, const int* __restrict__ mask,
    float* __restrict__ scores, int N, int E) {
  __shared__ float wLds[MAX_E];

#if __has_builtin(__builtin_amdgcn_tensor_load_to_lds)
  if (threadIdx.x < 32) {  // one wave issues the DMA (TDM ignores EXEC)
    const unsigned long long ga = (unsigned long long)(uintptr_t)w;
    const unsigned ldsAddr =
        (unsigned)(uintptr_t)(__attribute__((address_space(3))) float*)wLds;
    u32x4 g0;
    g0.x = 1u;                                   // count=1 valid descriptor
    g0.y = ldsAddr;                              // lds_addr [63:32]
    g0.z = (unsigned)(ga & 0xFFFFFFFFu);         // global_addr [95:64]
    g0.w = (unsigned)((ga >> 32) & 0x01FFFFFFu)  // global_addr [120:96]
           | (2u << 30);                         // type=2 ("image") [127:126]
    i32x8 g1;
    g1[0] = (int)(2u << 16);                     // data_size=4B; wg_mask=0
    g1[1] = (int)(((unsigned)E & 0xFFFFu) << 16);        // tensor_dim0 lo16
    g1[2] = (int)((((unsigned)E >> 16) & 0xFFFFu)        // tensor_dim0 hi16
                  | (1u << 16));                         // tensor_dim1 = 1
    g1[3] = (int)(((unsigned)E & 0xFFFFu) << 16);        // tile_dim0 = E
    g1[4] = 1;                                   // tile_dim1=1, tile_dim2=0
    g1[5] = E;                                   // tensor_dim0_stride lo32
    g1[6] = 0;
    g1[7] = 0;
    i32x4 gz = {0, 0, 0, 0};
    i32x8 gz8 = {0, 0, 0, 0, 0, 0, 0, 0};
    __builtin_amdgcn_tensor_load_to_lds(g0, g1, gz, gz, gz8, 0);
    __builtin_amdgcn_s_wait_tensorcnt(0);
  }
#else
  for (int i = threadIdx.x; i < E; i += blockDim.x) wLds[i] = w[i];
#endif
  __syncthreads();

  const int lane = threadIdx.x & 31;
  const int wave = blockIdx.x * (blockDim.x >> 5) + (threadIdx.x >> 5);
  const int row0 = wave * 16;
  if (row0 >= N) return;

  const int half = lane >> 4;  // 0: lanes 0-15, 1: lanes 16-31
  const int sub  = lane & 15;
  const float* rowPtr = emb + (size_t)(row0 + sub) * E + half * 2;

  v8f acc = {};
  for (int k = 0; k < E; k += 4) {
    const v2f a = *(const v2f*)(rowPtr + k);        // exact A-tile layout
    v2f bv;                                         // B[k'][n] = w[k'] ∀n
    bv.x = wLds[k + half * 2];
    bv.y = wLds[k + 1 + half * 2];
    acc = __builtin_amdgcn_wmma_f32_16x16x4_f32(false, a, false, bv,
                                                (short)0, acc, false, false);
  }

  // All 16 columns identical; lane with sub==0 in each half writes 8 rows.
  if (sub == 0) {
    const float bias = b[0];
#pragma unroll
    for (int m = 0; m < 8; ++m) {
      const int r = row0 + half * 8 + m;
      scores[r] = acc[m] + bias + __logf((float)mask[r]);
    }
  }
}

// ---------------------------------------------------------------------------
// Kernel 2: single-block bitonic sort of N=8192 packed keys in 64KB LDS.
// key = (~monotonic(score) << 32) | index  -> ascending sort == stable
// descending-score argsort (ties broken by ascending index, matching jnp).
// ---------------------------------------------------------------------------
__global__ __launch_bounds__(1024) void sp_sort_scores(
    const float* __restrict__ scores, int* __restrict__ order, int N) {
  __shared__ unsigned long long keys[8192];
  for (int i = threadIdx.x; i < N; i += 1024) {
    const unsigned u = __float_as_uint(scores[i]);
    const unsigned m = (u & 0x80000000u) ? ~u : (u | 0x80000000u);  // asc map
    keys[i] = ((unsigned long long)(~m) << 32) | (unsigned)i;       // descend
  }
  __syncthreads();
  for (int k = 2; k <= N; k <<= 1) {
    for (int j = k >> 1; j > 0; j >>= 1) {
      for (int i = threadIdx.x; i < N; i += 1024) {
        const int ixj = i ^ j;
        if (ixj > i) {
          const bool up = ((i & k) == 0);
          const unsigned long long a = keys[i], c = keys[ixj];
          if ((a > c) == up) { keys[i] = c; keys[ixj] = a; }
        }
      }
      __syncthreads();
    }
  }
  for (int i = threadIdx.x; i < N; i += 1024)
    order[i] = (int)(keys[i] & 0xFFFFFFFFu);
}

// ---------------------------------------------------------------------------
// Kernel 3: gather spans into sorted (score) order.
// ---------------------------------------------------------------------------
__global__ __launch_bounds__(256) void sp_gather_spans(
    const int* __restrict__ spans, const int* __restrict__ order,
    int* __restrict__ sS, int* __restrict__ sE, int N) {
  const int p = blockIdx.x * blockDim.x + threadIdx.x;
  if (p < N) {
    const int o = order[p];
    sS[p] = spans[2 * o];
    sE[p] = spans[2 * o + 1];
  }
}

// ---------------------------------------------------------------------------
// Kernel 4: keep[p] = 1 iff sorted span p overlaps no earlier sorted span.
// Tiled O(N^2) with the j-tile staged in LDS.
// ---------------------------------------------------------------------------
__global__ __launch_bounds__(256) void sp_keep(
    const int* __restrict__ sS, const int* __restrict__ sE,
    int* __restrict__ keep, int N) {
  __shared__ int tS[256], tE[256];
  const int p = blockIdx.x * 256 + threadIdx.x;
  int myS = 0, myE = 0;
  if (p < N) { myS = sS[p]; myE = sE[p]; }
  bool dead = false;
  const int limit = min(N, (int)(blockIdx.x + 1) * 256);
  for (int base = 0; base < limit; base += 256) {
    const int j = base + threadIdx.x;
    if (j < N) { tS[threadIdx.x] = sS[j]; tE[threadIdx.x] = sE[j]; }
    __syncthreads();
    const int tcount = min(256, limit - base);
    const int jmax = min(tcount, p - base);   // only j < p matter
    for (int t = 0; t < jmax; ++t) {
      const int js = tS[t], je = tE[t];
      dead |= ((myS <= js) & (js <= myE)) | ((js <= myS) & (myS <= je));
    }
    __syncthreads();
  }
  if (p < N) keep[p] = dead ? 0 : 1;
}

// ---------------------------------------------------------------------------
// Kernel 5: rank = keep ? prefix_kept : totalKept + prefix_notkept; select
// ranks < K, sort chosen original indices ascending (bitonic), emit small
// outputs. Single block of 1024 threads, 8 elements each (N = 8192).
// ---------------------------------------------------------------------------
__global__ __launch_bounds__(1024) void sp_select(
    const int* __restrict__ keep, const int* __restrict__ order,
    const float* __restrict__ scores, const int* __restrict__ mask,
    int* __restrict__ selIdx, float* __restrict__ outMask,
    float* __restrict__ outIdx, float* __restrict__ outScores,
    int N, int K) {
  __shared__ int tsum[1024];
  __shared__ int sel[1024];
  const int t = threadIdx.x;
  const int per = N >> 10;          // 8 for N=8192
  const int base = t * per;

  int loc[8];
  int s = 0;
#pragma unroll
  for (int q = 0; q < 8; ++q) {
    if (q < per) { loc[q] = s; s += keep[base + q]; }
  }
  tsum[t] = s;
  __syncthreads();
  for (int off = 1; off < 1024; off <<= 1) {     // Hillis-Steele inclusive
    const int v = tsum[t];
    const int add = (t >= off) ? tsum[t - off] : 0;
    __syncthreads();
    tsum[t] = v + add;
    __syncthreads();
  }
  const int totalKept = tsum[1023];
  const int exBefore = (t == 0) ? 0 : tsum[t - 1];
#pragma unroll
  for (int q = 0; q < 8; ++q) {
    if (q < per) {
      const int p = base + q;
      const int exK = exBefore + loc[q];
      const int r = keep[p] ? exK : totalKept + (p - exK);
      if (r < K) sel[r] = order[p];
    }
  }
  __syncthreads();
  // bitonic sort sel[0..K-1] ascending (K is a power of two, K <= 1024)
  for (int kk = 2; kk <= K; kk <<= 1) {
    for (int j = kk >> 1; j > 0; j >>= 1) {
      const int ixj = t ^ j;
      if (ixj > t && ixj < K) {
        const bool up = ((t & kk) == 0);
        const int a = sel[t], c = sel[ixj];
        if ((a > c) == up) { sel[t] = c; sel[ixj] = a; }
      }
      __syncthreads();
    }
  }
  if (t < K) {
    const int idx = sel[t];
    selIdx[t] = idx;
    outIdx[t] = (float)idx;
    outMask[t] = (float)mask[idx];
    outScores[t] = scores[idx];
  }
}

// ---------------------------------------------------------------------------
// Kernel 6: gather K embedding rows, float4-coalesced (one block per row).
// ---------------------------------------------------------------------------
__global__ __launch_bounds__(256) void sp_gather_emb(
    const float* __restrict__ emb, const int* __restrict__ selIdx,
    float* __restrict__ outEmb, int E) {
  const int k = blockIdx.x;
  const int src = selIdx[k];
  const float4* sp = (const float4*)(emb + (size_t)src * E);
  float4* dp = (float4*)(outEmb + (size_t)k * E);
  for (int c = threadIdx.x; c < (E >> 2); c += blockDim.x) dp[c] = sp[c];
}

// ---------------------------------------------------------------------------
extern "C" void kernel_launch(void* const* d_in, const int* in_sizes, int n_in,
                              void* d_out, int out_size, void* d_ws, size_t ws_size,
                              hipStream_t stream) {
  const float* emb   = (const float*)d_in[0];   // (1, N, E) f32
  const int*   spans = (const int*)d_in[1];     // (1, N, 2) i32
  const int*   mask  = (const int*)d_in[2];     // (1, N)    i32
  const float* w     = (const float*)d_in[3];   // (E, 1)    f32
  const float* b     = (const float*)d_in[4];   // (1,)      f32
  // d_in[5] = num_spans_to_keep lives on device; derive K from out_size.

  const int N = in_sizes[2];            // 8192
  const int E = in_sizes[0] / N;        // 1024
  const int K = out_size / (E + 3);     // 1024 (emb K*E + mask K + idx K + scores K)

  char* wsb = (char*)d_ws;
  float* scores = (float*)(wsb);                      // N f32
  int*   order  = (int*)(wsb + (size_t)N * 4);        // N i32
  int*   sS     = (int*)(wsb + (size_t)N * 8);        // N i32
  int*   sE     = (int*)(wsb + (size_t)N * 12);       // N i32
  int*   keep   = (int*)(wsb + (size_t)N * 16);       // N i32
  int*   selIdx = (int*)(wsb + (size_t)N * 20);       // K i32

  float* outEmb    = (float*)d_out;                   // K*E
  float* outMask   = outEmb + (size_t)K * E;          // K
  float* outIdx    = outMask + K;                     // K
  float* outScores = outIdx + K;                      // K

  const int waves = N / 16;                           // one wave per 16 rows
  sp_score_wmma<<<waves / 8, 256, 0, stream>>>(emb, w, b, mask, scores, N, E);
  sp_sort_scores<<<1, 1024, 0, stream>>>(scores, order, N);
  sp_gather_spans<<<(N + 255) / 256, 256, 0, stream>>>(spans, order, sS, sE, N);
  sp_keep<<<N / 256, 256, 0, stream>>>(sS, sE, keep, N);
  sp_select<<<1, 1024, 0, stream>>>(keep, order, scores, mask, selIdx,
                                    outMask, outIdx, outScores, N, K);
  sp_gather_emb<<<K, 256, 0, stream>>>(emb, selIdx, outEmb, E);
}